// CTCLayer_41034117546205
// MI455X (gfx1250) — compile-verified
//
#include <hip/hip_runtime.h>
#include <hip/hip_bf16.h>
#include <math.h>

// Problem constants (fixed by the reference setup_inputs()).
#define CTC_B 256
#define CTC_T 512
#define CTC_C 256
#define CTC_L 128
#define NEGV (-1.0e30f)
#define EPSV (1e-7f)
#define DEPTH 4  // prefetch pipeline depth (ring of row buffers)

// ---------------------------------------------------------------------------
// gfx1250 async global->LDS copy path (guarded; fallback = sync float4 copy)
// ---------------------------------------------------------------------------
#if defined(__HIP_DEVICE_COMPILE__) && __has_builtin(__builtin_amdgcn_global_load_async_to_lds_b128)
#define USE_ASYNC_LDS 1
#else
#define USE_ASYNC_LDS 0
#endif

#if USE_ASYNC_LDS
typedef int v4i __attribute__((vector_size(16)));
typedef __attribute__((address_space(1))) v4i* gv4_t;  // global, 16B vector
typedef __attribute__((address_space(3))) v4i* lv4_t;  // LDS, 16B vector
#if __has_builtin(__builtin_amdgcn_s_wait_asynccnt)
#define WAIT_ASYNC(n) __builtin_amdgcn_s_wait_asynccnt(n)
#else
#define WAIT_ASYNC(n) asm volatile("s_wait_asynccnt " #n ::: "memory")
#endif
#else
#define WAIT_ASYNC(n) ((void)0)
#endif

// Stage one [C]=256-float row of y_pred into LDS. 32 lanes x 16B x 2 = 1024B.
__device__ __forceinline__ void stage_row(const float* __restrict__ g,
                                          float* __restrict__ l, int lane) {
#if USE_ASYNC_LDS
  gv4_t gp = (gv4_t)((const char*)g + (size_t)lane * 16);
  lv4_t lp = (lv4_t)((char*)l + lane * 16);
  // Same IOFFSET is applied to both the global and LDS address (ISA 08 §4.4).
  __builtin_amdgcn_global_load_async_to_lds_b128(gp, lp, 0, 0);
  __builtin_amdgcn_global_load_async_to_lds_b128(gp, lp, 512, 0);
#else
  float4 v0 = ((const float4*)g)[lane];
  float4 v1 = ((const float4*)g)[lane + 32];
  ((float4*)l)[lane] = v0;
  ((float4*)l)[lane + 32] = v1;
#endif
}

// logaddexp with finite NEG sentinel (-1e30): exp(-huge)->0, log(1+0)->0.
// Fast path: e = exp(d) in (0,1], so log(1+e) argument is in [1,2] -> the
// cheap v_log_f32-based __logf loses at most ~6e-8 absolute per op.
__device__ __forceinline__ float lae(float a, float b) {
  float m = fmaxf(a, b);
  float d = fminf(a, b) - m;  // <= 0
  return m + __logf(1.0f + __expf(d));
}

__global__ void __launch_bounds__(32)
ctc_batch_cost_kernel(const int* __restrict__ y_true,
                      const float* __restrict__ y_pred,
                      float* __restrict__ out) {
  __shared__ float rows[DEPTH][CTC_C];  // ring of probability rows (4 KB)

  const int b = blockIdx.x;
  const int lane = threadIdx.x;  // wave32: one wave per batch row
  const float* base = y_pred + (size_t)b * CTC_T * CTC_C;
  const int* yb = y_true + (size_t)b * CTC_L;

  // Lane l owns extended states s in [8l, 8l+7]; odd states are labels
  // 4l..4l+3. Precompute label class ids and skip-transition legality.
  int lab[4];
  bool sk[4];
#pragma unroll
  for (int j = 0; j < 4; ++j) {
    int li = 4 * lane + j;
    lab[j] = yb[li];
    int pv = (li > 0) ? yb[li - 1] : -1;
    sk[j] = (li > 0) && (lab[j] != pv);
  }

  // Prime the pipeline: rows t = 0..DEPTH-1 in flight (2 async ops each).
#pragma unroll
  for (int t = 0; t < DEPTH; ++t) {
    stage_row(base + (size_t)t * CTC_C, rows[t], lane);
  }
  WAIT_ASYNC(6);  // <= 2*(DEPTH-1) outstanding => row 0 landed (in-order)

  // t = 0 init: alpha0[s] = (s < 2) ? lp[0][s] : NEG
  float lpb0 = __logf(rows[0][CTC_C - 1] + EPSV);
  float lpl0 = __logf(rows[0][lab[0]] + EPSV);
  float a0 = (lane == 0) ? lpb0 : NEGV;
  float a1 = (lane == 0) ? lpl0 : NEGV;
  float a2 = NEGV, a3 = NEGV, a4 = NEGV, a5 = NEGV;
  float a6 = NEGV, a7 = NEGV, a8 = NEGV;  // a8 = state 256 (lane 31 only)

#pragma unroll 4
  for (int t = 1; t < CTC_T; ++t) {
    // Keep DEPTH-1 rows ahead in flight; retire row t before computing.
    if (t + DEPTH - 1 < CTC_T) {
      stage_row(base + (size_t)(t + DEPTH - 1) * CTC_C,
                rows[(t + DEPTH - 1) % DEPTH], lane);
      WAIT_ASYNC(6);  // first of the 4 in-flight stages (row t) is done
    } else {
      WAIT_ASYNC(0);  // tail: drain everything outstanding
    }
    const float* row = rows[t % DEPTH];

    // Gather per-lane symbol probabilities from LDS; blank is a broadcast.
    float pb = row[CTC_C - 1];
    float p0 = row[lab[0]], p1 = row[lab[1]], p2 = row[lab[2]], p3 = row[lab[3]];
    float lpb = __logf(pb + EPSV);
    float q0 = __logf(p0 + EPSV);
    float q1 = __logf(p1 + EPSV);
    float q2 = __logf(p2 + EPSV);
    float q3 = __logf(p3 + EPSV);

    // Only cross-lane value needed: previous lane's a7 (= alpha[8l-1]).
    float pm1 = __shfl_up(a7, 1, 32);
    if (lane == 0) pm1 = NEGV;

    // Descending update so each new value reads only old values.
    float n8 = lpb + lae(a8, a7);                               // s=256 blank
    float n7 = q3 + lae(lae(a7, a6), sk[3] ? a5 : NEGV);        // label 4l+3
    float n6 = lpb + lae(a6, a5);                               // blank
    float n5 = q2 + lae(lae(a5, a4), sk[2] ? a3 : NEGV);        // label 4l+2
    float n4 = lpb + lae(a4, a3);                               // blank
    float n3 = q1 + lae(lae(a3, a2), sk[1] ? a1 : NEGV);        // label 4l+1
    float n2 = lpb + lae(a2, a1);                               // blank
    float n1 = q0 + lae(lae(a1, a0), sk[0] ? pm1 : NEGV);       // label 4l
    float n0 = lpb + lae(a0, pm1);                              // blank
    a0 = n0; a1 = n1; a2 = n2; a3 = n3; a4 = n4;
    a5 = n5; a6 = n6; a7 = n7; a8 = n8;
  }

  // loss = -logaddexp(alpha[S-1], alpha[S-2]) ; S-1=256 (a8@31), S-2=255 (a7@31)
  if (lane == 31) {
    out[b] = -lae(a8, a7);
  }
}

extern "C" void kernel_launch(void* const* d_in, const int* in_sizes, int n_in,
                              void* d_out, int out_size, void* d_ws, size_t ws_size,
                              hipStream_t stream) {
  const int* y_true = (const int*)d_in[0];     // [B, L] int32
  const float* y_pred = (const float*)d_in[1]; // [B, T, C] float32 (softmaxed)
  float* out = (float*)d_out;                  // [B, 1] float32
  (void)in_sizes; (void)n_in; (void)out_size; (void)d_ws; (void)ws_size;
  ctc_batch_cost_kernel<<<dim3(CTC_B), dim3(32), 0, stream>>>(y_true, y_pred, out);
}